// LocationRelativeAttention_28535762714639
// MI455X (gfx1250) — compile-verified
//
#include <hip/hip_runtime.h>
#include <hip/hip_bf16.h>
#include <math.h>

#define TLEN   4096
#define BATCH  64
#define HIDDEN 256
#define QDIM   1024
#define FILTER 31
#define PAD    15
#define WINDOW 1024
#define HCHUNK 32

typedef float v2f __attribute__((ext_vector_type(2)));
typedef float v8f __attribute__((ext_vector_type(8)));

// ---------------------------------------------------------------------------
// Kernel 1: q = query @ wq.T + bq via V_WMMA_F32_16X16X4_F32 (one wave/tile).
// A tile: 16 batches x 4 K ; B tile: 4 K x 16 H ; C: 16x16 f32.
// Also zeroes the score scratch buffer (d_ws is poisoned by the harness).
// ---------------------------------------------------------------------------
__global__ __launch_bounds__(32) void qproj_wmma_kernel(
    const float* __restrict__ query,   // [B, QDIM]
    const float* __restrict__ wq,      // [H, QDIM]
    const float* __restrict__ bq,      // [H]
    float* __restrict__ qbuf,          // [B, H] scratch out
    float* __restrict__ score)         // [B, WINDOW] scratch, zeroed here
{
    const int lane = threadIdx.x;            // 0..31, wave32
    const int m    = lane & 15;              // row within 16
    const int kp   = lane >> 4;              // K pair select (0/1)
    const int b0   = blockIdx.x * 16;        // batch tile
    const int h0   = blockIdx.y * 16;        // hidden tile

    v8f c = {};
    const float* aptr = query + (size_t)(b0 + m) * QDIM + 2 * kp;
    const float* bptr = wq    + (size_t)(h0 + m) * QDIM + 2 * kp;
    for (int k0 = 0; k0 < QDIM; k0 += 4) {
        // A-matrix 16x4 f32: VGPR0 holds K=0 (lanes 0-15) / K=2 (lanes 16-31),
        // VGPR1 holds K=1 / K=3  ->  per-lane v2f = A[m][2*kp + {0,1}]
        v2f a = *(const v2f*)(aptr + k0);
        // B-matrix 4x16: mirrored striping, per-lane v2f = B[2*kp+{0,1}][n]
        // where B[k][n] = wq[h0+n][k]
        v2f bm = *(const v2f*)(bptr + k0);
        c = __builtin_amdgcn_wmma_f32_16x16x4_f32(
                /*neg_a=*/false, a, /*neg_b=*/false, bm,
                /*c_mod=*/(short)0, c, /*reuse_a=*/false, /*reuse_b=*/false);
    }

    // D layout: VGPR r -> M = r + 8*kp (lanes>=16), N = lane&15
#pragma unroll
    for (int r = 0; r < 8; ++r) {
        const int bb = b0 + r + 8 * kp;
        const int hh = h0 + m;
        qbuf[bb * HIDDEN + hh] = c[r] + bq[hh];
    }

    // Zero the score accumulator buffer (grid = 4x16 blocks x 32 lanes = 2048)
    const int tg = (blockIdx.y * gridDim.x + blockIdx.x) * 32 + lane;
    const int nt = gridDim.x * gridDim.y * 32;
    for (int i = tg; i < BATCH * WINDOW; i += nt) score[i] = 0.0f;
}

// ---------------------------------------------------------------------------
// Kernel 2: partial scores. grid (B, H/HCHUNK). Each block: 32 channels over
// all 1024 window positions; conv31 + tanh + ws-weighted sum; atomic combine.
// ---------------------------------------------------------------------------
__global__ __launch_bounds__(256) void score_partial_kernel(
    const float* __restrict__ ca,            // [B, T]
    const unsigned char* __restrict__ mask,  // [B, T] bool
    const float* __restrict__ init_ca,       // [B, 1]
    const int* __restrict__ window_start,    // [B]
    const float* __restrict__ conv_w,        // [H, 1, FILTER]
    const float* __restrict__ conv_b,        // [H]
    const float* __restrict__ qbuf,          // [B, H]
    const float* __restrict__ wsv,           // [H]
    float* __restrict__ score)               // [B, WINDOW] accumulated
{
    __shared__ float s_loc[WINDOW + 2 * PAD];
    __shared__ float s_cw[HCHUNK * FILTER];
    __shared__ float s_bias[HCHUNK];
    __shared__ float s_wsv[HCHUNK];

    const int b   = blockIdx.x;
    const int h0  = blockIdx.y * HCHUNK;
    const int tid = threadIdx.x;
    const int wsb = window_start[b];
    const float iv = init_ca[b];

    // Stage the padded alignment window: loc = [init*pad | ca_masked | 0*pad]
    for (int j = tid; j < WINDOW + 2 * PAD; j += 256) {
        const int p = wsb + j;
        float v;
        if (p < PAD) {
            v = iv;
        } else if (p < PAD + TLEN) {
            const int t = p - PAD;
            v = mask[b * TLEN + t] ? ca[b * TLEN + t] : 0.0f;
        } else {
            v = 0.0f;
        }
        s_loc[j] = v;
    }
    for (int j = tid; j < HCHUNK * FILTER; j += 256)
        s_cw[j] = conv_w[h0 * FILTER + j];
    if (tid < HCHUNK) {
        s_bias[tid] = conv_b[h0 + tid] + qbuf[b * HIDDEN + h0 + tid];
        s_wsv[tid]  = wsv[h0 + tid];
    }
    __syncthreads();

    float acc[4] = {0.0f, 0.0f, 0.0f, 0.0f};
    for (int h = 0; h < HCHUNK; ++h) {
        const float wsh  = s_wsv[h];
        const float bias = s_bias[h];
        const float* cw  = &s_cw[h * FILTER];
#pragma unroll
        for (int wi = 0; wi < 4; ++wi) {
            const int w = tid + wi * 256;
            float s = bias;
#pragma unroll
            for (int k = 0; k < FILTER; ++k)
                s = fmaf(cw[k], s_loc[w + k], s);
            acc[wi] += wsh * tanhf(s);
        }
    }
#pragma unroll
    for (int wi = 0; wi < 4; ++wi)
        atomicAdd(&score[b * WINDOW + tid + wi * 256], acc[wi]);
}

// ---------------------------------------------------------------------------
// Kernel 3: per-batch softmax, context gather (BW hotspot: 67 MB of tokens),
// scatter alignment to [B,T], cum update, first-occurrence argmax -> new_ws.
// ---------------------------------------------------------------------------
__global__ __launch_bounds__(256) void finalize_kernel(
    const float* __restrict__ tokens,        // [T, B, H]
    const unsigned char* __restrict__ mask,  // [B, T]
    const int* __restrict__ num_tokens,      // [B]
    const float* __restrict__ ca,            // [B, T]
    const int* __restrict__ window_start,    // [B]
    const float* __restrict__ score,         // [B, WINDOW]
    float* __restrict__ out_ctx,             // [B, H]
    float* __restrict__ out_cum,             // [B, T]
    float* __restrict__ out_align,           // [B, T]
    float* __restrict__ out_newws)           // [B] (as float)
{
    __shared__ float s_align[WINDOW];
    __shared__ float s_red[256];
    __shared__ int   s_ri[256];

    const int b   = blockIdx.x;
    const int tid = threadIdx.x;
    const int wsb = window_start[b];

    // masked scores + local max
    float sc[4];
    float lmax = -INFINITY;
#pragma unroll
    for (int wi = 0; wi < 4; ++wi) {
        const int w = tid + wi * 256;
        float s = score[b * WINDOW + w];
        if (!mask[b * TLEN + wsb + w]) s = -INFINITY;
        sc[wi] = s;
        lmax = fmaxf(lmax, s);
    }
    s_red[tid] = lmax;
    __syncthreads();
    for (int off = 128; off > 0; off >>= 1) {
        if (tid < off) s_red[tid] = fmaxf(s_red[tid], s_red[tid + off]);
        __syncthreads();
    }
    const float mx = s_red[0];
    __syncthreads();

    float lsum = 0.0f;
#pragma unroll
    for (int wi = 0; wi < 4; ++wi) {
        sc[wi] = expf(sc[wi] - mx);
        lsum += sc[wi];
    }
    s_red[tid] = lsum;
    __syncthreads();
    for (int off = 128; off > 0; off >>= 1) {
        if (tid < off) s_red[tid] += s_red[tid + off];
        __syncthreads();
    }
    const float inv = 1.0f / s_red[0];
    __syncthreads();

#pragma unroll
    for (int wi = 0; wi < 4; ++wi)
        s_align[tid + wi * 256] = sc[wi] * inv;
    __syncthreads();

    // context[b, tid] = sum_w align[w] * tokens[wsb+w, b, tid]  (coalesced)
    const size_t rowstride = (size_t)BATCH * HIDDEN;
    const float* tok = tokens + (size_t)wsb * rowstride + (size_t)b * HIDDEN + tid;
    float ctx = 0.0f;
    for (int w = 0; w < WINDOW; w += 4) {
        if (w + 32 < WINDOW)
            __builtin_prefetch(tok + (size_t)(w + 32) * rowstride, 0, 0);
#pragma unroll
        for (int u = 0; u < 4; ++u)
            ctx = fmaf(s_align[w + u], tok[(size_t)(w + u) * rowstride], ctx);
    }
    out_ctx[b * HIDDEN + tid] = ctx;

    // scatter alignment + cumulative update over full T
    for (int t = tid; t < TLEN; t += 256) {
        const int w = t - wsb;
        const float af  = (w >= 0 && w < WINDOW) ? s_align[w] : 0.0f;
        const float cam = mask[b * TLEN + t] ? ca[b * TLEN + t] : 0.0f;
        out_align[b * TLEN + t] = af;
        out_cum[b * TLEN + t]   = cam + af;
    }

    // first-occurrence argmax over window (outside window is 0 < softmax max)
    float bv = -INFINITY;
    int   bi = 0x7fffffff;
#pragma unroll
    for (int wi = 0; wi < 4; ++wi) {
        const int w = tid + wi * 256;
        const float v = s_align[w];
        if (v > bv || (v == bv && w < bi)) { bv = v; bi = w; }
    }
    s_red[tid] = bv;
    s_ri[tid]  = bi;
    __syncthreads();
    for (int off = 128; off > 0; off >>= 1) {
        if (tid < off) {
            const float ov = s_red[tid + off];
            const int   oi = s_ri[tid + off];
            if (ov > s_red[tid] || (ov == s_red[tid] && oi < s_ri[tid])) {
                s_red[tid] = ov;
                s_ri[tid]  = oi;
            }
        }
        __syncthreads();
    }
    if (tid == 0) {
        const int tstar = wsb + s_ri[0];
        int nw = tstar - WINDOW / 2;
        const int cap = num_tokens[b] - WINDOW;
        nw = (nw < cap) ? nw : cap;
        nw = (nw > 0) ? nw : 0;
        out_newws[b] = (float)nw;
    }
}

extern "C" void kernel_launch(void* const* d_in, const int* in_sizes, int n_in,
                              void* d_out, int out_size, void* d_ws, size_t ws_size,
                              hipStream_t stream) {
    const float*         tokens      = (const float*)d_in[0];
    const unsigned char* tokens_mask = (const unsigned char*)d_in[1];
    const int*           num_tokens  = (const int*)d_in[2];
    const float*         query       = (const float*)d_in[3];   // [1,B,Q] -> [B,Q]
    const float*         ca          = (const float*)d_in[4];
    const float*         init_ca     = (const float*)d_in[5];
    const int*           win_start   = (const int*)d_in[6];
    const float*         conv_w      = (const float*)d_in[7];
    const float*         conv_b      = (const float*)d_in[8];
    const float*         wq          = (const float*)d_in[9];
    const float*         bq          = (const float*)d_in[10];
    const float*         wsv         = (const float*)d_in[11];

    float* out  = (float*)d_out;
    float* qbuf = (float*)d_ws;                   // [B, H]
    float* scor = qbuf + BATCH * HIDDEN;          // [B, WINDOW]

    float* out_ctx   = out;
    float* out_cum   = out + BATCH * HIDDEN;
    float* out_align = out + BATCH * HIDDEN + BATCH * TLEN;
    float* out_newws = out + BATCH * HIDDEN + 2 * BATCH * TLEN;

    qproj_wmma_kernel<<<dim3(BATCH / 16, HIDDEN / 16), 32, 0, stream>>>(
        query, wq, bq, qbuf, scor);

    score_partial_kernel<<<dim3(BATCH, HIDDEN / HCHUNK), 256, 0, stream>>>(
        ca, tokens_mask, init_ca, win_start, conv_w, conv_b, qbuf, wsv, scor);

    finalize_kernel<<<BATCH, 256, 0, stream>>>(
        tokens, tokens_mask, num_tokens, ca, win_start, scor,
        out_ctx, out_cum, out_align, out_newws);
}